// SPDNet_8426725834947
// MI455X (gfx1250) — compile-verified
//
#include <hip/hip_runtime.h>
#include <hip/hip_bf16.h>
#include <math.h>

// SPDNet on gfx1250: one wave32 per batch element.
//   S1 = W1^T x W1          (WMMA f32 16x16x4, padded 32x64x60 / 32x32x60)
//   eigh(S1) -> V1,w1 ; Y = W2^T V1 ; S2 = Y diag(max(w1,EPS)) Y^T   (WMMA)
//   eigh(S2) -> V2,w2 ; Z = W3^T V2 ; S3 = Z diag(max(w2,EPS)) Z^T   (scalar, 5x10)
//   eigh(S3) -> V3,w3 ; L = V3 diag(log w3) V3^T ; triu -> linear head

typedef __attribute__((ext_vector_type(2))) float v2f;
typedef __attribute__((ext_vector_type(8))) float v8f;

#define B_TOT 16384
#define NCLS  17
#define EPSV  1e-4f
#define NWAVES 4

static __device__ __forceinline__ v8f wmma4(v2f a, v2f b, v8f c) {
  return __builtin_amdgcn_wmma_f32_16x16x4_f32(false, a, false, b, (short)0, c,
                                               false, false);
}

// Cyclic Jacobi eigendecomposition of symmetric n x n (n<=20) matrix in LDS,
// row stride 32. V gets eigenvectors as columns. One wave, lockstep, no barriers.
static __device__ void eigh_jacobi(float* A, float* V, int n, int sweeps, int lane) {
  for (int t = lane; t < n * 32; t += 32) {
    int r = t >> 5, c = t & 31;
    V[t] = (r == c) ? 1.0f : 0.0f;   // zero pad cols too (read by WMMA B frags)
  }
  for (int s = 0; s < sweeps; ++s) {
    for (int p = 0; p < n - 1; ++p) {
      for (int q = p + 1; q < n; ++q) {
        float apq = A[p * 32 + q];            // wave-uniform broadcast read
        if (fabsf(apq) > 1e-12f) {            // wave-uniform branch
          float app = A[p * 32 + p];
          float aqq = A[q * 32 + q];
          float tau = (aqq - app) / (2.0f * apq);
          float tj  = (tau >= 0.0f ? 1.0f : -1.0f) /
                      (fabsf(tau) + sqrtf(1.0f + tau * tau));
          float c   = 1.0f / sqrtf(1.0f + tj * tj);
          float sn  = tj * c;
          int j = lane;
          bool act = (j < n) && (j != p) && (j != q);
          float rpj = 0.0f, rqj = 0.0f;
          if (act) { rpj = A[p * 32 + j]; rqj = A[q * 32 + j]; }
          if (act) {
            A[p * 32 + j] = c * rpj - sn * rqj;
            A[q * 32 + j] = sn * rpj + c * rqj;
          }
          float ajp = 0.0f, ajq = 0.0f;
          if (act) { ajp = A[j * 32 + p]; ajq = A[j * 32 + q]; }
          if (act) {
            A[j * 32 + p] = c * ajp - sn * ajq;
            A[j * 32 + q] = sn * ajp + c * ajq;
          }
          if (lane == 0) {
            A[p * 32 + p] = app - tj * apq;
            A[q * 32 + q] = aqq + tj * apq;
            A[p * 32 + q] = 0.0f;
            A[q * 32 + p] = 0.0f;
          }
          float vjp = 0.0f, vjq = 0.0f;
          if (j < n) { vjp = V[j * 32 + p]; vjq = V[j * 32 + q]; }
          if (j < n) {
            V[j * 32 + p] = c * vjp - sn * vjq;
            V[j * 32 + q] = sn * vjp + c * vjq;
          }
        }
      }
    }
  }
}

__global__ __launch_bounds__(32 * NWAVES) void spdnet_kernel(
    const float* __restrict__ x,  const float* __restrict__ W1,
    const float* __restrict__ W2, const float* __restrict__ W3,
    const float* __restrict__ linw, const float* __restrict__ linb,
    float* __restrict__ out) {
  // ---- shared weights (zero padded for WMMA tiles) ----
  __shared__ float W1s[60 * 32];        // [k][m], k<57 m<20 valid
  __shared__ float W2s[20 * 16];        // [k][m], k<20 m<10 valid
  __shared__ float W3s[10 * 5];
  __shared__ float linws[NCLS * 15];
  __shared__ float linbs[NCLS];
  // ---- per-wave scratch ----
  __shared__ float Us [NWAVES][20 * 60];  // U = W1^T x   (cols 57..59 zero)
  __shared__ float Ssh[NWAVES][20 * 32];  // current SPD matrix (stride 32)
  __shared__ float Vsh[NWAVES][20 * 32];  // eigenvectors   (stride 32)
  __shared__ float Ysh[NWAVES][16 * 20];  // Y = W2^T V1
  __shared__ float Zsh[NWAVES][5 * 10];   // Z = W3^T V2
  __shared__ float wcs[NWAVES][20];       // clamped/logged eigenvalues
  __shared__ float vcs[NWAVES][15];       // triu vector

  const int tid = threadIdx.x;

  for (int t = tid; t < 60 * 32; t += blockDim.x) W1s[t] = 0.0f;
  for (int t = tid; t < 20 * 16; t += blockDim.x) W2s[t] = 0.0f;
  __syncthreads();
  for (int t = tid; t < 57 * 20; t += blockDim.x) W1s[(t / 20) * 32 + (t % 20)] = W1[t];
  for (int t = tid; t < 20 * 10; t += blockDim.x) W2s[(t / 10) * 16 + (t % 10)] = W2[t];
  for (int t = tid; t < 10 * 5;  t += blockDim.x) W3s[t] = W3[t];
  for (int t = tid; t < NCLS * 15; t += blockDim.x) linws[t] = linw[t];
  for (int t = tid; t < NCLS;      t += blockDim.x) linbs[t] = linb[t];
  __syncthreads();

  const int wv   = tid >> 5;
  const int lane = tid & 31;
  const int b    = blockIdx.x * NWAVES + wv;
  float* U  = Us[wv];  float* S  = Ssh[wv]; float* V = Vsh[wv];
  float* Y  = Ysh[wv]; float* Z  = Zsh[wv];
  float* wc = wcs[wv]; float* vec = vcs[wv];
  const float* xg = x + (size_t)b * (57 * 57);

  const int lm   = lane & 15;            // m/n inside 16x16 tile
  const int lkb  = (lane >> 4) << 1;     // k base: 0 or 2
  const int sel8 = (lane >> 4) << 3;     // D row offset: 0 or 8

  // ---------- Stage 1: U = W1^T @ x   (M=20 N=57 K=57, padded 32/64/60) ----------
  for (int m0 = 0; m0 < 32; m0 += 16) {
    for (int n0 = 0; n0 < 64; n0 += 16) {
      v8f acc = {};
      for (int k0 = 0; k0 < 60; k0 += 4) {
        int kk = k0 + lkb;
        int nn = n0 + lm;
        v2f a, bb;
        a.x = W1s[kk * 32 + m0 + lm];
        a.y = W1s[(kk + 1) * 32 + m0 + lm];
        bb.x = ((kk     < 57) && (nn < 57)) ? xg[kk * 57 + nn]       : 0.0f;
        bb.y = ((kk + 1 < 57) && (nn < 57)) ? xg[(kk + 1) * 57 + nn] : 0.0f;
        acc = wmma4(a, bb, acc);
      }
#pragma unroll
      for (int r = 0; r < 8; ++r) {
        int row = m0 + r + sel8, col = n0 + lm;
        if (row < 20 && col < 60) U[row * 60 + col] = acc[r];
      }
    }
  }

  // ---------- Stage 2: S1 = U @ W1   (20x20) ----------
  for (int m0 = 0; m0 < 32; m0 += 16) {
    for (int n0 = 0; n0 < 32; n0 += 16) {
      v8f acc = {};
      for (int k0 = 0; k0 < 60; k0 += 4) {
        int kk = k0 + lkb;
        int mm = m0 + lm;
        v2f a, bb;
        a.x = (mm < 20) ? U[mm * 60 + kk]     : 0.0f;
        a.y = (mm < 20) ? U[mm * 60 + kk + 1] : 0.0f;
        bb.x = W1s[kk * 32 + n0 + lm];
        bb.y = W1s[(kk + 1) * 32 + n0 + lm];
        acc = wmma4(a, bb, acc);
      }
#pragma unroll
      for (int r = 0; r < 8; ++r) {
        int row = m0 + r + sel8, col = n0 + lm;
        if (row < 20 && col < 20) S[row * 32 + col] = acc[r];
      }
    }
  }

  // ---------- ReEig 1 ----------
  eigh_jacobi(S, V, 20, 10, lane);
  if (lane < 20) wc[lane] = fmaxf(S[lane * 32 + lane], EPSV);

  // ---------- Y = W2^T @ V1  (10x20, padded 16x32x20) ----------
  for (int n0 = 0; n0 < 32; n0 += 16) {
    v8f acc = {};
    for (int k0 = 0; k0 < 20; k0 += 4) {
      int kk = k0 + lkb;
      v2f a, bb;
      a.x = W2s[kk * 16 + lm];
      a.y = W2s[(kk + 1) * 16 + lm];
      bb.x = V[kk * 32 + n0 + lm];
      bb.y = V[(kk + 1) * 32 + n0 + lm];
      acc = wmma4(a, bb, acc);
    }
#pragma unroll
    for (int r = 0; r < 8; ++r) {
      int row = r + sel8, col = n0 + lm;
      if (col < 20) Y[row * 20 + col] = acc[r];
    }
  }

  // ---------- S2 = Y diag(wc) Y^T  (10x10, one tile, K=20) ----------
  {
    v8f acc = {};
    for (int k0 = 0; k0 < 20; k0 += 4) {
      int kk = k0 + lkb;
      v2f a, bb;
      float y0 = Y[lm * 20 + kk], y1 = Y[lm * 20 + kk + 1];
      a.x = y0 * wc[kk];
      a.y = y1 * wc[kk + 1];
      bb.x = y0;            // B[k][n] = Y[n][k]
      bb.y = y1;
      acc = wmma4(a, bb, acc);
    }
#pragma unroll
    for (int r = 0; r < 8; ++r) S[(r + sel8) * 32 + lm] = acc[r];
  }

  // ---------- ReEig 2 ----------
  eigh_jacobi(S, V, 10, 8, lane);
  if (lane < 10) wc[lane] = fmaxf(S[lane * 32 + lane], EPSV);

  // ---------- Z = W3^T @ V2  (5x10, scalar) ----------
  for (int t = lane; t < 50; t += 32) {
    int i = t / 10, j = t % 10;
    float acc = 0.0f;
    for (int k = 0; k < 10; ++k) acc += W3s[k * 5 + i] * V[k * 32 + j];
    Z[i * 10 + j] = acc;
  }
  // ---------- S3 = Z diag(wc) Z^T  (5x5) ----------
  if (lane < 25) {
    int i = lane / 5, j = lane % 5;
    float acc = 0.0f;
    for (int k = 0; k < 10; ++k) acc += Z[i * 10 + k] * wc[k] * Z[j * 10 + k];
    S[i * 32 + j] = acc;
  }

  // ---------- LogEig ----------
  eigh_jacobi(S, V, 5, 7, lane);
  if (lane < 5) wc[lane] = logf(S[lane * 32 + lane]);

  // ---------- triu vectorization: vec[t] = L[i][j], L = V diag(log w) V^T ----------
  if (lane < 15) {
    int i = (lane >= 5) + (lane >= 9) + (lane >= 12) + (lane >= 14);
    int st = (i == 0) ? 0 : (i == 1) ? 5 : (i == 2) ? 9 : (i == 3) ? 12 : 14;
    int j = lane - st + i;
    float acc = 0.0f;
    for (int k = 0; k < 5; ++k) acc += V[i * 32 + k] * wc[k] * V[j * 32 + k];
    vec[lane] = acc;
  }

  // ---------- linear head ----------
  if (lane < NCLS) {
    float acc = linbs[lane];
    for (int t = 0; t < 15; ++t) acc += vec[t] * linws[lane * 15 + t];
    out[(size_t)b * NCLS + lane] = acc;
  }
}

extern "C" void kernel_launch(void* const* d_in, const int* in_sizes, int n_in,
                              void* d_out, int out_size, void* d_ws, size_t ws_size,
                              hipStream_t stream) {
  const float* x    = (const float*)d_in[0];
  const float* W1   = (const float*)d_in[1];
  const float* W2   = (const float*)d_in[2];
  const float* W3   = (const float*)d_in[3];
  const float* linw = (const float*)d_in[4];
  const float* linb = (const float*)d_in[5];
  float* out = (float*)d_out;

  dim3 grid(B_TOT / NWAVES);
  dim3 block(32 * NWAVES);
  hipLaunchKernelGGL(spdnet_kernel, grid, block, 0, stream,
                     x, W1, W2, W3, linw, linb, out);
}